// HeteroClassifier_9345848836757
// MI455X (gfx1250) — compile-verified
//
#include <hip/hip_runtime.h>
#include <math.h>

typedef float v2f __attribute__((ext_vector_type(2)));
typedef float v8f __attribute__((ext_vector_type(8)));

#define BATCH_C 1024

// ---------------- utility ----------------
__global__ void k_zero(float* __restrict__ p, long n) {
  long i = (long)blockIdx.x * blockDim.x + threadIdx.x;
  if (i < n) p[i] = 0.0f;
}

// Fused weight products: TW2[512,32] = trans_w @ gat_w ; tb2[32] = trans_b @ gat_w
__global__ void k_proj_w(const float* __restrict__ trans_w, const float* __restrict__ trans_b,
                         const float* __restrict__ gat_w, float* __restrict__ TW2,
                         float* __restrict__ tb2) {
  int idx = blockIdx.x * blockDim.x + threadIdx.x;
  if (idx >= 513 * 32) return;
  int r = idx >> 5, c = idx & 31;
  float acc = 0.f;
  for (int k = 0; k < 50; ++k) {
    float a = (r < 512) ? trans_w[r * 50 + k] : trans_b[k];
    acc = fmaf(a, gat_w[k * 32 + c], acc);
  }
  if (r < 512) TW2[r * 32 + c] = acc; else tb2[c] = acc;
}

// ---------------- f32 WMMA GEMM: C[M,32] = A[M,K](lda) @ B[K,32] (+bias) ----------------
// One wave computes a 16x16 output tile via V_WMMA_F32_16X16X4_F32.
// A 16x4 layout: lane = m + 16*(k>>1), vgpr = k&1. B 4x16 mirrored. D: 8 VGPRs per ISA.
// K is a compile-time constant (and even) so the main loop carries NO bounds checks:
// A fragment = one aligned 8-byte vector load, B fragment = two b32 loads.
template <int K>
__global__ void k_gemm32(const float* __restrict__ A, int lda,
                         const float* __restrict__ Bm,
                         const float* __restrict__ bias,
                         float* __restrict__ C, int M) {
  static_assert((K & 1) == 0, "K must be even");
  int tid  = threadIdx.x;
  int lane = tid & 31;
  int wave = blockIdx.x * (blockDim.x >> 5) + (tid >> 5);
  int nWaves = (M >> 4) * 2;            // N=32 -> 2 col tiles
  if (wave >= nWaves) return;           // wave-uniform: EXEC all-1s inside
  int tm = wave >> 1, tn = wave & 1;
  int mloc = lane & 15, khalf = lane >> 4;
  int arow = tm * 16 + mloc;
  int ncol = tn * 16 + mloc;
  const long abase = (long)arow * lda;  // even + even k0 -> 8B aligned v2f loads
  v8f acc = {};

  constexpr int KMAIN = K & ~3;
#pragma unroll 4
  for (int kb = 0; kb < KMAIN; kb += 4) {
    int k0 = kb + khalf * 2;
    v2f a = *(const v2f*)(A + abase + k0);
    v2f b;
    b.x = Bm[k0 * 32 + ncol];
    b.y = Bm[(k0 + 1) * 32 + ncol];
    acc = __builtin_amdgcn_wmma_f32_16x16x4_f32(false, a, false, b, (short)0, acc,
                                                false, false);
  }
  if constexpr ((K & 3) != 0) {
    // statically-known remainder (K=50 -> 2 extra k's); branchless:
    // clamp address in-bounds (even), zero invalid A lanes with cndmask (a*b -> 0).
    int k0 = KMAIN + khalf * 2;
    int k0c = (k0 < K) ? k0 : (K - 2);
    v2f a = *(const v2f*)(A + abase + k0c);
    float valid = (k0 < K) ? 1.f : 0.f;  // K even, k0 even: k0<K <=> k0+1<K
    a.x *= valid;
    a.y *= valid;
    v2f b;
    b.x = Bm[k0c * 32 + ncol];
    b.y = Bm[(k0c + 1) * 32 + ncol];
    acc = __builtin_amdgcn_wmma_f32_16x16x4_f32(false, a, false, b, (short)0, acc,
                                                false, false);
  }

  float bv = bias ? bias[ncol] : 0.f;
#pragma unroll
  for (int r = 0; r < 8; ++r) {
    int om = tm * 16 + r + khalf * 8;   // D layout: lanes16-31 hold M = r+8
    C[(long)om * 32 + ncol] = acc[r] + bv;
  }
}

// z rows [0, NW) = zw[word_ids]  (gather of precomputed wordemb@gat_w)
__global__ void k_gather_words(const int* __restrict__ wids, const float* __restrict__ zw,
                               float* __restrict__ z, int NW) {
  long i = (long)blockIdx.x * blockDim.x + threadIdx.x;
  if (i >= (long)NW * 32) return;
  int n = (int)(i >> 5), c = (int)(i & 31);
  z[i] = zw[(long)wids[n] * 32 + c];
}

// el = z @ attn_l, er = z @ attn_r : one wave per node
__global__ void k_elr(const float* __restrict__ z, const float* __restrict__ al,
                      const float* __restrict__ ar, float* __restrict__ el,
                      float* __restrict__ er, int n) {
  int node = blockIdx.x * 8 + (threadIdx.x >> 5);
  if (node >= n) return;
  int c = threadIdx.x & 31;
  float v = z[(long)node * 32 + c];
  float l = v * al[c], r = v * ar[c];
  for (int off = 16; off > 0; off >>= 1) {
    l += __shfl_xor(l, off, 32);
    r += __shfl_xor(r, off, 32);
  }
  if (c == 0) { el[node] = l; er[node] = r; }
}

// pass 1: e = leaky_relu(el[src]+er[dst], 0.2); per-dst max via ordered-uint atomicMax
__global__ void k_edge1(const int* __restrict__ src, const int* __restrict__ dst,
                        const float* __restrict__ el, const float* __restrict__ er,
                        float* __restrict__ e_arr, unsigned* __restrict__ mkeys, int E) {
  int e = blockIdx.x * blockDim.x + threadIdx.x;
  if (e >= E) return;
  int s = src[e], d = dst[e];
  float t = el[s] + er[d];
  float ev = (t >= 0.f) ? t : 0.2f * t;
  e_arr[e] = ev;
  unsigned b = __float_as_uint(ev);
  unsigned key = (b & 0x80000000u) ? ~b : (b | 0x80000000u);  // order-preserving map
  atomicMax(&mkeys[d], key);
}

// pass 2: alpha = exp(e - m[dst]); hsum[dst] += alpha*z[src]; denom[dst] += alpha
// one wave per edge (lane = channel): contiguous 128B gather + contiguous atomics
__global__ void k_edge2(const int* __restrict__ src, const int* __restrict__ dst,
                        const float* __restrict__ e_arr, const unsigned* __restrict__ mkeys,
                        const float* __restrict__ z, float* __restrict__ denom,
                        float* __restrict__ hsum, int E) {
  int e = blockIdx.x * 8 + (threadIdx.x >> 5);
  if (e >= E) return;
  int lane = threadIdx.x & 31;
  int s = src[e], d = dst[e];
  unsigned k = mkeys[d];
  unsigned b = (k & 0x80000000u) ? (k & 0x7FFFFFFFu) : ~k;
  float m = __uint_as_float(b);
  float alpha = expf(e_arr[e] - m);
  atomicAdd(&hsum[(long)d * 32 + lane], alpha * z[(long)s * 32 + lane]);
  if (lane == 0) atomicAdd(&denom[d], alpha);
}

// h = hsum/denom + gat_b ; scatter into per-graph sums (one wave per node)
__global__ void k_node(const float* __restrict__ hsum, const float* __restrict__ denom,
                       const float* __restrict__ gat_b, const int* __restrict__ gid,
                       float* __restrict__ gsum, float* __restrict__ cnt, int n) {
  int node = blockIdx.x * 8 + (threadIdx.x >> 5);
  if (node >= n) return;
  int c = threadIdx.x & 31;
  float h = hsum[(long)node * 32 + c] / denom[node] + gat_b[c];
  int g = gid[node];
  atomicAdd(&gsum[g * 32 + c], h);
  if (c == 0) atomicAdd(&cnt[g], 1.0f);
}

__global__ void k_gmean(const float* __restrict__ gsum, const float* __restrict__ cnt,
                        float* __restrict__ gemb) {
  int i = blockIdx.x * blockDim.x + threadIdx.x;
  if (i >= BATCH_C * 32) return;
  gemb[i] = gsum[i] / fmaxf(cnt[i >> 5], 1.0f);
}

// s1[i] = sigmoid( relu(A1[i]+P2[i]) . W2 + b2 )   (one wave per row)
__global__ void k_s1(const float* __restrict__ A1, const float* __restrict__ P2,
                     const float* __restrict__ W2, const float* __restrict__ b2,
                     float* __restrict__ s1) {
  int i = blockIdx.x * 8 + (threadIdx.x >> 5);
  if (i >= BATCH_C) return;
  int c = threadIdx.x & 31;
  float y = fmaxf(A1[i * 32 + c] + P2[i * 32 + c], 0.f);
  float v = y * W2[c];
  for (int off = 16; off > 0; off >>= 1) v += __shfl_xor(v, off, 32);
  if (c == 0) s1[i] = 1.f / (1.f + expf(-(v + b2[0])));
}

// s2[i] = max_j sigmoid( relu(A1[i]+A2[j]) . W2 + b2 ) * click[i,j]   (block per row)
__global__ void k_s2(const float* __restrict__ A1, const float* __restrict__ A2,
                     const float* __restrict__ W2, const float* __restrict__ b2,
                     const int* __restrict__ click, float* __restrict__ s2) {
  __shared__ float sA1[32], sW2[32], red[256];
  int i = blockIdx.x, t = threadIdx.x;
  if (t < 32) { sA1[t] = A1[i * 32 + t]; sW2[t] = W2[t]; }
  __syncthreads();
  float bb = b2[0];
  float mx = 0.f;                       // res >= 0 always (sigmoid>0, click in {0,1})
  for (int j = t; j < BATCH_C; j += 256) {
    float acc = 0.f;
    const float* a2 = A2 + j * 32;
#pragma unroll
    for (int c = 0; c < 32; ++c)
      acc = fmaf(fmaxf(sA1[c] + a2[c], 0.f), sW2[c], acc);
    float sg = 1.f / (1.f + expf(-(acc + bb)));
    mx = fmaxf(mx, sg * (float)click[i * BATCH_C + j]);
  }
  red[t] = mx;
  __syncthreads();
  for (int off = 128; off > 0; off >>= 1) {
    if (t < off) red[t] = fmaxf(red[t], red[t + off]);
    __syncthreads();
  }
  if (t == 0) s2[i] = red[0];
}

__global__ void k_loss(const float* __restrict__ s1, const float* __restrict__ s2,
                       float* __restrict__ out) {
  __shared__ float r1[256], r2[256];
  int t = threadIdx.x;
  float a = 0.f, b = 0.f;
  for (int i = t; i < BATCH_C; i += 256) {
    a += logf(s1[i] + 1e-12f);
    b += logf(1.f - s2[i] + 1e-12f);
  }
  r1[t] = a; r2[t] = b;
  __syncthreads();
  for (int off = 128; off > 0; off >>= 1) {
    if (t < off) { r1[t] += r1[t + off]; r2[t] += r2[t + off]; }
    __syncthreads();
  }
  if (t == 0) out[0] = -(r1[0] / (float)BATCH_C) - (r2[0] / (float)BATCH_C);
}

// ---------------- host launch ----------------
extern "C" void kernel_launch(void* const* d_in, const int* in_sizes, int n_in,
                              void* d_out, int out_size, void* d_ws, size_t ws_size,
                              hipStream_t stream) {
  const int*   word_ids_a = (const int*)d_in[0];
  const int*   word_ids_p = (const int*)d_in[1];
  const float* img_a      = (const float*)d_in[2];
  const float* img_p      = (const float*)d_in[3];
  const int*   src        = (const int*)d_in[4];
  const int*   dst        = (const int*)d_in[5];
  const int*   gid        = (const int*)d_in[6];
  const int*   click      = (const int*)d_in[7];
  const float* wordemb    = (const float*)d_in[8];
  const float* trans_w    = (const float*)d_in[9];
  const float* trans_b    = (const float*)d_in[10];
  const float* gat_w      = (const float*)d_in[11];
  const float* attn_l     = (const float*)d_in[12];
  const float* attn_r     = (const float*)d_in[13];
  const float* gat_b      = (const float*)d_in[14];
  const float* W1         = (const float*)d_in[15];
  const float* b1         = (const float*)d_in[16];
  const float* W2         = (const float*)d_in[17];
  const float* b2         = (const float*)d_in[18];

  const int NW = in_sizes[0];            // 400000
  const int NI = in_sizes[2] / 512;      // 50000
  const int Nn = in_sizes[6];            // 450000
  const int E  = in_sizes[4];            // 3450000
  const int VB = in_sizes[8] / 50;       // 50000

  // workspace layout (floats, 64-elem aligned)
  float* ws = (float*)d_ws;
  size_t o = 0;
  auto alloc = [&](size_t n) { size_t r = o; o += (n + 63) & ~(size_t)63; return r; };
  size_t oTW2  = alloc(512 * 32);
  size_t oTB2  = alloc(32);
  size_t oZW   = alloc((size_t)VB * 32);
  size_t oZ    = alloc((size_t)Nn * 32);
  size_t oEL   = alloc(Nn);
  size_t oER   = alloc(Nn);
  size_t oEARR = alloc(E);
  size_t oMK   = alloc(Nn);
  size_t oDEN  = alloc(Nn);
  size_t oHS   = alloc((size_t)Nn * 32);
  size_t oGS   = alloc(BATCH_C * 32);
  size_t oCNT  = alloc(BATCH_C);
  size_t oGA   = alloc(BATCH_C * 32);
  size_t oGP   = alloc(BATCH_C * 32);
  size_t oA1   = alloc(BATCH_C * 32);
  size_t oA2   = alloc(BATCH_C * 32);
  size_t oP2   = alloc(BATCH_C * 32);
  size_t oS1   = alloc(BATCH_C);
  size_t oS2   = alloc(BATCH_C);
  (void)ws_size; (void)out_size; (void)n_in;

  auto g1 = [](long n) { return (unsigned)((n + 255) / 256); };

  // shared precompute: TW2/tb2, zw = wordemb @ gat_w (WMMA, K=50)
  k_proj_w<<<g1(513 * 32), 256, 0, stream>>>(trans_w, trans_b, gat_w, ws + oTW2, ws + oTB2);
  {
    int waves = (VB >> 4) * 2;
    k_gemm32<50><<<(waves + 3) / 4, 128, 0, stream>>>(wordemb, 50, gat_w, nullptr,
                                                      ws + oZW, VB);
  }

  // one graph-embedding pass
  auto embed = [&](const int* wids, const float* img, float* gemb) {
    k_zero<<<g1(Nn), 256, 0, stream>>>(ws + oMK, Nn);          // mkeys=0 (< any key)
    k_zero<<<g1(Nn), 256, 0, stream>>>(ws + oDEN, Nn);
    k_zero<<<g1((long)Nn * 32), 256, 0, stream>>>(ws + oHS, (long)Nn * 32);
    k_zero<<<g1(BATCH_C * 32), 256, 0, stream>>>(ws + oGS, BATCH_C * 32);
    k_zero<<<g1(BATCH_C), 256, 0, stream>>>(ws + oCNT, BATCH_C);

    int waves = (NI >> 4) * 2;
    k_gemm32<512><<<(waves + 3) / 4, 128, 0, stream>>>(img, 512, ws + oTW2, ws + oTB2,
                                                       ws + oZ + (size_t)NW * 32, NI);
    k_gather_words<<<g1((long)NW * 32), 256, 0, stream>>>(wids, ws + oZW, ws + oZ, NW);
    k_elr<<<(Nn + 7) / 8, 256, 0, stream>>>(ws + oZ, attn_l, attn_r, ws + oEL, ws + oER, Nn);
    k_edge1<<<g1(E), 256, 0, stream>>>(src, dst, ws + oEL, ws + oER,
                                       ws + oEARR, (unsigned*)(ws + oMK), E);
    k_edge2<<<(E + 7) / 8, 256, 0, stream>>>(src, dst, ws + oEARR, (const unsigned*)(ws + oMK),
                                             ws + oZ, ws + oDEN, ws + oHS, E);
    k_node<<<(Nn + 7) / 8, 256, 0, stream>>>(ws + oHS, ws + oDEN, gat_b, gid,
                                             ws + oGS, ws + oCNT, Nn);
    k_gmean<<<g1(BATCH_C * 32), 256, 0, stream>>>(ws + oGS, ws + oCNT, gemb);
  };

  embed(word_ids_a, img_a, ws + oGA);
  embed(word_ids_p, img_p, ws + oGP);

  // discriminator GEMMs (WMMA, M=1024, K=32)
  {
    int waves = (BATCH_C >> 4) * 2;
    unsigned gb = (waves + 3) / 4;
    k_gemm32<32><<<gb, 128, 0, stream>>>(ws + oGA, 32, W1,           b1,      ws + oA1, BATCH_C);
    k_gemm32<32><<<gb, 128, 0, stream>>>(ws + oGA, 32, W1 + 32 * 32, nullptr, ws + oA2, BATCH_C);
    k_gemm32<32><<<gb, 128, 0, stream>>>(ws + oGP, 32, W1 + 32 * 32, nullptr, ws + oP2, BATCH_C);
  }
  k_s1<<<(BATCH_C + 7) / 8, 256, 0, stream>>>(ws + oA1, ws + oP2, W2, b2, ws + oS1);
  k_s2<<<BATCH_C, 256, 0, stream>>>(ws + oA1, ws + oA2, W2, b2, click, ws + oS2);
  k_loss<<<1, 256, 0, stream>>>(ws + oS1, ws + oS2, (float*)d_out);
}